// AcousticPropagator_43757126812023
// MI455X (gfx1250) — compile-verified
//
#include <hip/hip_runtime.h>
#include <hip/hip_bf16.h>

// MI455X / gfx1250 acoustic FDTD propagator.
//  * fused v+p update per step (halo recompute), double-buffered state in d_ws
//  * LDS tiles staged with CDNA5 async global->LDS (global_load_async_to_lds_b32
//    + s_wait_asynccnt, ASYNCcnt)
//  * finite differences evaluated on the matrix pipes with
//    V_WMMA_F32_16X16X4_F32: diff = banded {-1,+1} matrix product; every output
//    is an f32 sum of exactly two nonzero terms -> bit-comparable to v_sub_f32,
//    and the FLOP inflation is free (problem is latency-bound: 1.6 GFLOP total,
//    3 MB state resident in the 192 MB L2).
// Workspace: 6*NS*NX*NZ + 3*NX*NZ floats ~= 6.8 MB.

#define NXD 256
#define NZD 256
#define NTD 400
#define NSD 4
#define NRD 128
#define DXD 10.0f
#define DTD 0.001f
#define TILE 32
#define N2D (NXD * NZD)
#define N3D (NSD * N2D)

typedef __attribute__((ext_vector_type(2))) float v2f;
typedef __attribute__((ext_vector_type(8))) float v8f;

__device__ __forceinline__ void async_ld_f32(unsigned lds_off, const float* g) {
  asm volatile("global_load_async_to_lds_b32 %0, %1, off"
               :: "v"(lds_off), "v"(g) : "memory");
}
__device__ __forceinline__ void wait_async_all() {
  asm volatile("s_wait_asynccnt 0" ::: "memory");
}
__device__ __forceinline__ unsigned lds_off(const void* p) {
  return (unsigned)(size_t)p;  // low 32 bits of generic LDS pointer = LDS offset
}

// D[i][j] = m[(r0+i+1)*stride + cb + j] - m[(r0+i)*stride + cb + j], 16x16.
// D = (-I+U) * M[r0..r0+15][cb..] + e_{15,0} * M[r0+16..][cb..]
__device__ __forceinline__ v8f wmma_rowdiff(const float* m, int stride, int r0, int cb) {
  const int lane = threadIdx.x & 31;
  const int n = lane & 15;
  const int half = lane >> 4;
  v8f acc = {};
#pragma unroll
  for (int k0 = 0; k0 < 16; k0 += 4) {
    const int ka = k0 + 2 * half;  // this lane's K slots: ka, ka+1
    v2f a, b;
    a.x = (ka == n) ? -1.0f : ((ka == n + 1) ? 1.0f : 0.0f);
    a.y = (ka + 1 == n) ? -1.0f : ((ka + 1 == n + 1) ? 1.0f : 0.0f);
    b.x = m[(r0 + ka) * stride + cb + n];
    b.y = m[(r0 + ka + 1) * stride + cb + n];
    acc = __builtin_amdgcn_wmma_f32_16x16x4_f32(false, a, false, b, (short)0,
                                                acc, false, false);
  }
  {  // boundary row r0+16 (only K=0 column of E nonzero; over-reads hit x0)
    const int ka = 2 * half;
    v2f a, b;
    a.x = (n == 15 && ka == 0) ? 1.0f : 0.0f;
    a.y = 0.0f;
    b.x = m[(r0 + 16 + ka) * stride + cb + n];
    b.y = m[(r0 + 16 + ka + 1) * stride + cb + n];
    acc = __builtin_amdgcn_wmma_f32_16x16x4_f32(false, a, false, b, (short)0,
                                                acc, false, false);
  }
  return acc;
}

// D[i][j] = m[(rb+i)*stride + c0 + j + 1] - m[(rb+i)*stride + c0 + j], 16x16.
// D = M[rb..][c0..c0+15] * C + M[rb..][c0+16..] * C2,  data is the A operand.
__device__ __forceinline__ v8f wmma_coldiff(const float* m, int stride, int rb, int c0) {
  const int lane = threadIdx.x & 31;
  const int n = lane & 15;
  const int half = lane >> 4;
  v8f acc = {};
#pragma unroll
  for (int k0 = 0; k0 < 16; k0 += 4) {
    const int ka = k0 + 2 * half;
    v2f a, b;
    a.x = m[(rb + n) * stride + c0 + ka];
    a.y = m[(rb + n) * stride + c0 + ka + 1];
    b.x = (ka == n) ? -1.0f : ((ka == n + 1) ? 1.0f : 0.0f);
    b.y = (ka + 1 == n) ? -1.0f : ((ka + 1 == n + 1) ? 1.0f : 0.0f);
    acc = __builtin_amdgcn_wmma_f32_16x16x4_f32(false, a, false, b, (short)0,
                                                acc, false, false);
  }
  {  // boundary column c0+16 via C2[0][15]=1 (garbage cols multiplied by 0)
    const int ka = 2 * half;
    v2f a, b;
    a.x = m[(rb + n) * stride + c0 + 16 + ka];
    a.y = m[(rb + n) * stride + c0 + 16 + ka + 1];
    b.x = (ka == 0 && n == 15) ? 1.0f : 0.0f;
    b.y = 0.0f;
    acc = __builtin_amdgcn_wmma_f32_16x16x4_f32(false, a, false, b, (short)0,
                                                acc, false, false);
  }
  return acc;
}

__global__ void coeff_k(const float* __restrict__ vp, const float* __restrict__ rho,
                        const float* __restrict__ damp, float* __restrict__ fac,
                        float* __restrict__ bsc, float* __restrict__ ksc) {
  int i = blockIdx.x * blockDim.x + threadIdx.x;
  if (i < N2D) {
    float r = rho[i], v = vp[i];
    fac[i] = 1.0f - DTD * damp[i];
    bsc[i] = DTD / (r * DXD);          // (DT/rho)/DX
    ksc[i] = DTD * (r * v * v) / DXD;  // DT*kappa/DX  (DX == DZ)
  }
}

__global__ void zero_k(float* __restrict__ a, int n) {
  int i = blockIdx.x * blockDim.x + threadIdx.x;
  if (i < n) a[i] = 0.0f;
}

__global__ __launch_bounds__(256) void step_k(
    const float* __restrict__ p_in, const float* __restrict__ vx_in,
    const float* __restrict__ vz_in, float* __restrict__ p_out,
    float* __restrict__ vx_out, float* __restrict__ vz_out,
    const float* __restrict__ fac, const float* __restrict__ bsc,
    const float* __restrict__ ksc, const float* __restrict__ wavelet,
    const int* __restrict__ src_x, const int* __restrict__ src_z,
    const int* __restrict__ rcv_x, const int* __restrict__ rcv_z,
    float* __restrict__ rec, int t) {
  __shared__ float sh_p[34 * 34];   // p + 1-cell halo, stride 34
  __shared__ float sh_vx[33 * 32];  // rows x0-1..x0+31, z interior, stride 32
  __shared__ float sh_vz[32 * 33];  // x interior, cols z0-1..z0+31, stride 33

  const int tid = threadIdx.x;
  const int wid = tid >> 5;
  const int lane = tid & 31;
  const int n = lane & 15;
  const int half = lane >> 4;
  const int s = blockIdx.z;
  const int x0 = blockIdx.x * TILE;
  const int z0 = blockIdx.y * TILE;
  const int base3 = s * N2D;

  // ---- async stage state tiles into LDS (clamped halo; garbage cells unused)
  const float* pb = p_in + base3;
  for (int i = tid; i < 34 * 34; i += 256) {
    int lx = i / 34, lz = i - lx * 34;
    int gx = min(max(x0 - 1 + lx, 0), NXD - 1);
    int gz = min(max(z0 - 1 + lz, 0), NZD - 1);
    async_ld_f32(lds_off(&sh_p[i]), pb + gx * NZD + gz);
  }
  const float* vxb = vx_in + base3;
  for (int i = tid; i < 33 * 32; i += 256) {
    int xi = i >> 5, lz = i & 31;
    int gx = max(x0 - 1 + xi, 0);
    async_ld_f32(lds_off(&sh_vx[i]), vxb + gx * NZD + (z0 + lz));
  }
  const float* vzb = vz_in + base3;
  for (int i = tid; i < 32 * 33; i += 256) {
    int xi = i / 33, lz = i - xi * 33;
    int gz = max(z0 - 1 + lz, 0);
    async_ld_f32(lds_off(&sh_vz[i]), vzb + (x0 + xi) * NZD + gz);
  }
  wait_async_all();
  __syncthreads();

  // ---- velocity phase: 8 waves, one 16x16 WMMA derivative block each
  if (wid < 4) {  // dpx blocks: xi in [r0,r0+16), lz in [c0,c0+16)
    const int r0 = (wid >> 1) << 4, c0 = (wid & 1) << 4;
    v8f d = wmma_rowdiff(sh_p, 34, r0, c0 + 1);
#pragma unroll
    for (int v = 0; v < 8; ++v) {
      int xi = r0 + v + 8 * half;  // C/D layout: VGPR v -> rows v / v+8
      int lz = c0 + n;
      int gx = x0 - 1 + xi;        // never reaches NXD-1 (leftover row does)
      int gz = z0 + lz;
      int ci = max(gx, 0) * NZD + gz;
      float vn = fac[ci] * sh_vx[xi * 32 + lz] - bsc[ci] * d[v];
      sh_vx[xi * 32 + lz] = vn;
      if (xi > 0) vx_out[base3 + gx * NZD + gz] = vn;
    }
  } else {  // dpz blocks
    const int w2 = wid - 4;
    const int r0 = (w2 >> 1) << 4, c0 = (w2 & 1) << 4;
    v8f d = wmma_coldiff(sh_p, 34, r0 + 1, c0);
#pragma unroll
    for (int v = 0; v < 8; ++v) {
      int xi = r0 + v + 8 * half;
      int lz = c0 + n;
      int gx = x0 + xi;
      int gz = z0 - 1 + lz;        // never reaches NZD-1 inside blocks
      int ci = gx * NZD + max(gz, 0);
      float vn = fac[ci] * sh_vz[xi * 33 + lz] - bsc[ci] * d[v];
      sh_vz[xi * 33 + lz] = vn;
      if (lz > 0) vz_out[base3 + gx * NZD + gz] = vn;
    }
  }
  // leftover halo lines (only cells that can touch the zero-derivative edge)
  if (wid == 0) {  // vx row xi = 32 (gx = x0+31)
    int lz = lane, gx = x0 + 31, gz = z0 + lz;
    float dpx = (gx < NXD - 1)
                    ? (sh_p[33 * 34 + lz + 1] - sh_p[32 * 34 + lz + 1]) : 0.0f;
    int ci = gx * NZD + gz;
    float vn = fac[ci] * sh_vx[32 * 32 + lz] - bsc[ci] * dpx;
    sh_vx[32 * 32 + lz] = vn;
    vx_out[base3 + ci] = vn;
  }
  if (wid == 1) {  // vz col lz = 32 (gz = z0+31)
    int xi = lane, gx = x0 + xi, gz = z0 + 31;
    float dpz = (gz < NZD - 1)
                    ? (sh_p[(xi + 1) * 34 + 33] - sh_p[(xi + 1) * 34 + 32]) : 0.0f;
    int ci = gx * NZD + gz;
    float vn = fac[ci] * sh_vz[xi * 33 + 32] - bsc[ci] * dpz;
    sh_vz[xi * 33 + 32] = vn;
    vz_out[base3 + ci] = vn;
  }
  __syncthreads();

  // ---- pressure phase: waves 0..3, one 16x16 block each (dvx + dvz WMMAs)
  if (wid < 4) {
    const int r0 = (wid >> 1) << 4, c0 = (wid & 1) << 4;
    v8f dx = wmma_rowdiff(sh_vx, 32, r0, c0);
    v8f dz = wmma_coldiff(sh_vz, 33, r0, c0);
    const int sx = src_x[s], sz = src_z[s];
    const float srcv = wavelet[s * NTD + t] * (DTD / (DXD * DXD));
    const int rz0 = rcv_z[0];
#pragma unroll
    for (int v = 0; v < 8; ++v) {
      int lx = r0 + v + 8 * half;
      int lz = c0 + n;
      int gx = x0 + lx, gz = z0 + lz;
      int ci = gx * NZD + gz;
      float dvx = (gx > 0) ? dx[v] : 0.0f;  // jnp.pad: zero backward diff at 0
      float dvz = (gz > 0) ? dz[v] : 0.0f;
      float pn = fac[ci] * sh_p[(lx + 1) * 34 + lz + 1] - ksc[ci] * (dvx + dvz);
      if (gx == sx && gz == sz) pn += srcv;
      p_out[base3 + ci] = pn;
      if (gz == rz0) {
        for (int r = 0; r < NRD; ++r)
          if (rcv_x[r] == gx && rcv_z[r] == gz)
            rec[(s * NTD + t) * NRD + r] = pn;
      }
    }
  }
}

extern "C" void kernel_launch(void* const* d_in, const int* in_sizes, int n_in,
                              void* d_out, int out_size, void* d_ws, size_t ws_size,
                              hipStream_t stream) {
  const float* vp = (const float*)d_in[0];
  const float* rho = (const float*)d_in[1];
  const float* damp = (const float*)d_in[2];
  const float* wav = (const float*)d_in[3];
  const int* src_x = (const int*)d_in[4];
  const int* src_z = (const int*)d_in[5];
  const int* rcv_x = (const int*)d_in[6];
  const int* rcv_z = (const int*)d_in[7];
  float* rec = (float*)d_out;

  float* ws = (float*)d_ws;
  float* pB[2] = {ws + 0 * N3D, ws + 3 * N3D};
  float* vxB[2] = {ws + 1 * N3D, ws + 4 * N3D};
  float* vzB[2] = {ws + 2 * N3D, ws + 5 * N3D};
  float* fac = ws + 6 * N3D;
  float* bsc = fac + N2D;
  float* ksc = bsc + N2D;

  // buffer 0 (p,vx,vz contiguous) must be zero at t=0; buffer 1 is fully
  // overwritten at t=0 before being read -> deterministic across replays.
  zero_k<<<(3 * N3D + 255) / 256, 256, 0, stream>>>(ws, 3 * N3D);
  coeff_k<<<(N2D + 255) / 256, 256, 0, stream>>>(vp, rho, damp, fac, bsc, ksc);

  dim3 grid(NXD / TILE, NZD / TILE, NSD);
  for (int t = 0; t < NTD; ++t) {
    int a = t & 1, b = a ^ 1;
    step_k<<<grid, 256, 0, stream>>>(pB[a], vxB[a], vzB[a], pB[b], vxB[b],
                                     vzB[b], fac, bsc, ksc, wav, src_x, src_z,
                                     rcv_x, rcv_z, rec, t);
  }
}